// AdExternal_N3Tree_14817637171540
// MI455X (gfx1250) — compile-verified
//
#include <hip/hip_runtime.h>
#include <math.h>

// ---------------------------------------------------------------------------
// N3Tree forward, restructured:
//   * scan groups are independent (group g only touches dat[g]); leaf cells
//     are never written by the scan, so output = MLP(leaf_original + acc)
//   * acc = sum over all nodes of depth_weight[d] * feat  (order-free sum)
//   * MLPs fused into one GEMM pair driven by V_WMMA_F32_16X16X4_F32
// ---------------------------------------------------------------------------

#define N_NODES   32768
#define DDIM      32
#define NGROUPS   4096          // parents 0..4095
#define ROWS      (N_NODES * 8) // 262144 flat cells
#define HID       128           // concat hidden: f(64) + s(64)

typedef float v2f __attribute__((ext_vector_type(2)));
typedef float v8f __attribute__((ext_vector_type(8)));

// workspace layout (float offsets); total 137648 floats = 550592 bytes
#define OFF_ACC   0                         // 32
#define OFF_DAT0  32                        // 256 (final dat[0] for root step)
#define OFF_PART  288                       // 4096*32 per-group partials
#define OFF_W1    (288 + NGROUPS * 32)      // 32*128 concatenated W1
#define OFF_B1    (OFF_W1 + 32 * HID)       // 128
#define OFF_W2    (OFF_B1 + HID)            // 128*16 block-diagonal W2 (padded)
#define OFF_B2    (OFF_W2 + HID * 16)       // 16

__device__ __forceinline__ float gelu_exact(float x) {
    return 0.5f * x * (1.0f + erff(x * 0.70710678118654752f));
}

// ---------------------------------------------------------------------------
// Kernel 0: pack MLP weights into fused layouts in ws
// ---------------------------------------------------------------------------
__global__ void prep_kernel(const float* __restrict__ fw1, const float* __restrict__ fb1,
                            const float* __restrict__ fw2, const float* __restrict__ fb2,
                            const float* __restrict__ sw1, const float* __restrict__ sb1,
                            const float* __restrict__ sw2, const float* __restrict__ sb2,
                            float* __restrict__ ws) {
    const int t = threadIdx.x;
    for (int idx = t; idx < 32 * HID; idx += blockDim.x) {
        int i = idx / HID, j = idx % HID;
        ws[OFF_W1 + idx] = (j < 64) ? fw1[i * 64 + j] : sw1[i * 64 + (j - 64)];
    }
    for (int j = t; j < HID; j += blockDim.x)
        ws[OFF_B1 + j] = (j < 64) ? fb1[j] : sb1[j - 64];
    for (int idx = t; idx < HID * 16; idx += blockDim.x) {
        int k = idx / 16, j = idx % 16;
        float v = 0.0f;
        if (k < 64 && j < 3)       v = fw2[k * 3 + j];
        else if (k >= 64 && j == 3) v = sw2[k - 64];
        ws[OFF_W2 + idx] = v;
    }
    for (int j = t; j < 16; j += blockDim.x) {
        float v = 0.0f;
        if (j < 3) v = fb2[j]; else if (j == 3) v = sb2[0];
        ws[OFF_B2 + j] = v;
    }
}

// ---------------------------------------------------------------------------
// Kernel 1: per-group sequential TreeConv chain. One wave (32 lanes) per
// group; lane = output feature. W row cached in padded LDS (stride 260 ->
// float4-aligned, bank-spread); x kept transposed (xT[i][k]) for broadcast
// float4 reads.
// ---------------------------------------------------------------------------
__global__ void scan_kernel(const float* __restrict__ data,
                            const float* __restrict__ convw,
                            const float* __restrict__ convb,
                            const float* __restrict__ dwt,
                            const int*   __restrict__ node_depth,
                            float* __restrict__ ws) {
    __shared__ float wW[32 * 260];   // 33280 B
    __shared__ float xT[32 * 8];     // xT[i*8+k] = node[k][i]
    const int g    = blockIdx.x;
    const int lane = threadIdx.x;    // 0..31
    const int d    = node_depth[8 * g + 1];

    // stage conv_w[d] rows (row o = 256 consecutive floats), coalesced
    const float* Wd = convw + d * 8192;
    for (int idx = lane; idx < 8192; idx += 32)
        wW[(idx >> 8) * 260 + (idx & 255)] = Wd[idx];
    // stage x transposed
    for (int k = 0; k < 8; ++k)
        xT[lane * 8 + k] = data[g * 256 + k * 32 + lane];
    const float bias = convb[d * 32 + lane];
    float psum = 0.0f;
    __syncthreads();

    for (int c = 7; c >= 0; --c) {
        const bool active = (8 * g + c + 1) < N_NODES;  // last group has 7 kids
        float f = bias;
        #pragma unroll 4
        for (int i = 0; i < 32; ++i) {
            const float4 w0 = *(const float4*)&wW[lane * 260 + i * 8];
            const float4 w1 = *(const float4*)&wW[lane * 260 + i * 8 + 4];
            const float4 x0 = *(const float4*)&xT[i * 8];        // broadcast
            const float4 x1 = *(const float4*)&xT[i * 8 + 4];
            f += w0.x * x0.x + w0.y * x0.y + w0.z * x0.z + w0.w * x0.w
               + w1.x * x1.x + w1.y * x1.y + w1.z * x1.z + w1.w * x1.w;
        }
        __syncthreads();                 // all reads before sibling write
        if (active) {
            xT[lane * 8 + c] = f;        // this node's feat replaces its cell
            psum += f;
        }
        __syncthreads();
    }
    ws[OFF_PART + g * 32 + lane] = dwt[d] * psum;
    if (g == 0) {                        // keep final dat[0] for the root step
        for (int k = 0; k < 8; ++k)
            ws[OFF_DAT0 + k * 32 + lane] = xT[lane * 8 + k];
    }
}

// ---------------------------------------------------------------------------
// Kernel 2: deterministic reduction of partials + root-node (depth 0) step.
// ---------------------------------------------------------------------------
__global__ void reduce_kernel(const float* __restrict__ convw,
                              const float* __restrict__ convb,
                              const float* __restrict__ dwt,
                              float* __restrict__ ws) {
    const int o = threadIdx.x;           // 0..31
    float a = 0.0f;
    for (int g = 0; g < NGROUPS; ++g)
        a += ws[OFF_PART + g * 32 + o];
    // node 0: feat = conv_w[0] applied to final dat[0]
    float f = convb[o];
    for (int i = 0; i < 32; ++i)
        for (int k = 0; k < 8; ++k)
            f += convw[o * 256 + i * 8 + k] * ws[OFF_DAT0 + k * 32 + i];
    a += dwt[0] * f;
    ws[OFF_ACC + o] = a;
}

// ---------------------------------------------------------------------------
// Kernel 3: fused leaf MLP via V_WMMA_F32_16X16X4_F32.
// One wave per 16-row tile; 4 waves/block. h staged in LDS (stride 132:
// 8B-aligned v2f, conflict-free A-operand reads). Non-leaf rows -> 0.
// ---------------------------------------------------------------------------
__global__ void mlp_kernel(const float* __restrict__ data,
                           const float* __restrict__ ws,
                           float* __restrict__ out) {
    __shared__ float hbuf[4 * 16 * 132];   // 33792 B
    const int lane = threadIdx.x & 31;
    const int wave = threadIdx.x >> 5;
    const int tile = blockIdx.x * 4 + wave;
    const int r0   = tile * 16;
    const int m15  = lane & 15;            // N for B/C operands, M for A
    const int hi   = lane >> 4;
    const int k0   = hi * 2;               // per-lane K pair within a k-block
    float* h = hbuf + wave * 16 * 132;

    const float* W1  = ws + OFF_W1;
    const float* B1  = ws + OFF_B1;
    const float* W2  = ws + OFF_W2;
    const float* B2  = ws + OFF_B2;
    const float* acc = ws + OFF_ACC;

    // A operand: x = leaf_row + acc, as 8 K-slices of 16x4
    v2f a[8];
    const int row = r0 + m15;
    #pragma unroll
    for (int kb = 0; kb < 8; ++kb) {
        const int col = kb * 4 + k0;
        const v2f dv = *(const v2f*)&data[row * 32 + col];
        a[kb].x = dv.x + acc[col];
        a[kb].y = dv.y + acc[col + 1];
    }

    // GEMM1: h(16x128) = x(16x32) @ W1cat(32x128), bias, exact GELU
    #pragma unroll
    for (int nb = 0; nb < 8; ++nb) {
        v8f c = {0.f, 0.f, 0.f, 0.f, 0.f, 0.f, 0.f, 0.f};
        #pragma unroll
        for (int kb = 0; kb < 8; ++kb) {
            const int kk = kb * 4 + k0;
            v2f b;
            b.x = W1[kk * HID + nb * 16 + m15];
            b.y = W1[(kk + 1) * HID + nb * 16 + m15];
            c = __builtin_amdgcn_wmma_f32_16x16x4_f32(false, a[kb], false, b,
                                                      (short)0, c, false, false);
        }
        const float bs = B1[nb * 16 + m15];
        #pragma unroll
        for (int v = 0; v < 8; ++v)
            h[(v + 8 * hi) * 132 + nb * 16 + m15] = gelu_exact(c[v] + bs);
    }
    __syncthreads();

    // GEMM2: out(16x16) = h(16x128) @ W2p(128x16)  (only cols 0..3 live)
    v8f c2 = {0.f, 0.f, 0.f, 0.f, 0.f, 0.f, 0.f, 0.f};
    #pragma unroll
    for (int kb = 0; kb < 32; ++kb) {
        const int kk = kb * 4 + k0;
        const v2f a2 = *(const v2f*)&h[m15 * 132 + kk];
        v2f b2;
        b2.x = W2[kk * 16 + m15];
        b2.y = W2[(kk + 1) * 16 + m15];
        c2 = __builtin_amdgcn_wmma_f32_16x16x4_f32(false, a2, false, b2,
                                                   (short)0, c2, false, false);
    }

    if (m15 < 4) {
        const float b2s = B2[m15];
        #pragma unroll
        for (int v = 0; v < 8; ++v) {
            const int rr = r0 + v + 8 * hi;
            const float val = (rr >= N_NODES - 1) ? (c2[v] + b2s) : 0.0f;
            out[rr * 4 + m15] = val;
        }
    }
}

// ---------------------------------------------------------------------------
extern "C" void kernel_launch(void* const* d_in, const int* in_sizes, int n_in,
                              void* d_out, int out_size, void* d_ws, size_t ws_size,
                              hipStream_t stream) {
    (void)in_sizes; (void)n_in; (void)out_size; (void)ws_size;
    const float* data  = (const float*)d_in[0];
    const float* convw = (const float*)d_in[1];
    const float* convb = (const float*)d_in[2];
    const float* dwt   = (const float*)d_in[3];
    const float* fw1   = (const float*)d_in[4];
    const float* fb1   = (const float*)d_in[5];
    const float* fw2   = (const float*)d_in[6];
    const float* fb2   = (const float*)d_in[7];
    const float* sw1   = (const float*)d_in[8];
    const float* sb1   = (const float*)d_in[9];
    const float* sw2   = (const float*)d_in[10];
    const float* sb2   = (const float*)d_in[11];
    const int*   ndep  = (const int*)d_in[13];
    float* wsf = (float*)d_ws;
    float* outp = (float*)d_out;

    prep_kernel  <<<1,       256, 0, stream>>>(fw1, fb1, fw2, fb2, sw1, sb1, sw2, sb2, wsf);
    scan_kernel  <<<NGROUPS,  32, 0, stream>>>(data, convw, convb, dwt, ndep, wsf);
    reduce_kernel<<<1,        32, 0, stream>>>(convw, convb, dwt, wsf);
    mlp_kernel   <<<ROWS / 64, 128, 0, stream>>>(data, wsf, outp);
}